// HardGouraudShader_16406775070836
// MI455X (gfx1250) — compile-verified
//
#include <hip/hip_runtime.h>
#include <math.h>

// Problem constants (from reference): V,F,N,H,W,K = 100000,200000,4,512,512,8
#define V_N   100000
#define F_N   200000
#define NPIX  (4 * 512 * 512)   // 1,048,576 pixels
#define K_N   8
#define EPSN  1e-6f

typedef int v4i __attribute__((ext_vector_type(4)));
typedef __attribute__((address_space(1))) int* as1_i32;
typedef __attribute__((address_space(3))) int* as3_i32;
typedef __attribute__((address_space(1))) v4i* as1_v4i;
typedef __attribute__((address_space(3))) v4i* as3_v4i;

// ---------------------------------------------------------------------------
// Kernel 0: zero the vertex-normal accumulator (workspace is NOT re-zeroed by
// the harness between replays, so we must do it every launch).
// ---------------------------------------------------------------------------
__global__ __launch_bounds__(256) void zero_kernel(float* __restrict__ p, int n) {
    int i = blockIdx.x * 256 + threadIdx.x;
    if (i < n) p[i] = 0.0f;
}

// ---------------------------------------------------------------------------
// Kernel 1: face normals, scatter-add into vertex normals (global_atomic_add_f32)
// ---------------------------------------------------------------------------
__global__ __launch_bounds__(256) void face_normals_kernel(
    const float* __restrict__ verts, const int* __restrict__ faces,
    float* __restrict__ vn)
{
    int f = blockIdx.x * 256 + threadIdx.x;
    if (f >= F_N) return;
    int i0 = faces[3 * f + 0], i1 = faces[3 * f + 1], i2 = faces[3 * f + 2];
    float ax = verts[3 * i0 + 0], ay = verts[3 * i0 + 1], az = verts[3 * i0 + 2];
    float bx = verts[3 * i1 + 0], by = verts[3 * i1 + 1], bz = verts[3 * i1 + 2];
    float cx = verts[3 * i2 + 0], cy = verts[3 * i2 + 1], cz = verts[3 * i2 + 2];
    float e1x = bx - ax, e1y = by - ay, e1z = bz - az;
    float e2x = cx - ax, e2y = cy - ay, e2z = cz - az;
    float nx = e1y * e2z - e1z * e2y;
    float ny = e1z * e2x - e1x * e2z;
    float nz = e1x * e2y - e1y * e2x;
    atomicAdd(&vn[3 * i0 + 0], nx); atomicAdd(&vn[3 * i0 + 1], ny); atomicAdd(&vn[3 * i0 + 2], nz);
    atomicAdd(&vn[3 * i1 + 0], nx); atomicAdd(&vn[3 * i1 + 1], ny); atomicAdd(&vn[3 * i1 + 2], nz);
    atomicAdd(&vn[3 * i2 + 0], nx); atomicAdd(&vn[3 * i2 + 1], ny); atomicAdd(&vn[3 * i2 + 2], nz);
}

// ---------------------------------------------------------------------------
// Kernel 2: per-vertex Gouraud shade -> vcol (V x 3)
// ---------------------------------------------------------------------------
__global__ __launch_bounds__(256) void shade_vertices_kernel(
    const float* __restrict__ verts, const float* __restrict__ vert_colors,
    const float* __restrict__ vn,
    const float* __restrict__ light_pos, const float* __restrict__ cam_pos,
    const float* __restrict__ amb, const float* __restrict__ dif,
    const float* __restrict__ spc,
    float* __restrict__ vcol)
{
    int v = blockIdx.x * 256 + threadIdx.x;
    if (v >= V_N) return;

    float nx = vn[3 * v + 0], ny = vn[3 * v + 1], nz = vn[3 * v + 2];
    float nn = sqrtf(nx * nx + ny * ny + nz * nz);
    float inv = 1.0f / fmaxf(nn, EPSN);
    nx *= inv; ny *= inv; nz *= inv;

    float px = verts[3 * v + 0], py = verts[3 * v + 1], pz = verts[3 * v + 2];

    float lx = light_pos[0] - px, ly = light_pos[1] - py, lz = light_pos[2] - pz;
    float ln = sqrtf(lx * lx + ly * ly + lz * lz);
    float li = 1.0f / fmaxf(ln, EPSN);
    lx *= li; ly *= li; lz *= li;

    float cosang = nx * lx + ny * ly + nz * lz;
    float crelu  = fmaxf(cosang, 0.0f);

    float wx = cam_pos[0] - px, wy = cam_pos[1] - py, wz = cam_pos[2] - pz;
    float wn = sqrtf(wx * wx + wy * wy + wz * wz);
    float wi = 1.0f / fmaxf(wn, EPSN);
    wx *= wi; wy *= wi; wz *= wi;

    float rx = -lx + 2.0f * cosang * nx;
    float ry = -ly + 2.0f * cosang * ny;
    float rz = -lz + 2.0f * cosang * nz;
    float sc = fmaxf(rx * wx + ry * wy + rz * wz, 0.0f);
    // sc^64 via 6 squarings (SHININESS = 64)
    float s2 = sc * sc, s4 = s2 * s2, s8 = s4 * s4;
    float s16 = s8 * s8, s32 = s16 * s16, s64 = s32 * s32;
    float specf = (cosang > 0.0f) ? s64 : 0.0f;

    float c0 = vert_colors[3 * v + 0], c1 = vert_colors[3 * v + 1], c2 = vert_colors[3 * v + 2];
    vcol[3 * v + 0] = (amb[0] + dif[0] * crelu) * c0 + spc[0] * specf;
    vcol[3 * v + 1] = (amb[1] + dif[1] * crelu) * c1 + spc[1] * specf;
    vcol[3 * v + 2] = (amb[2] + dif[2] * crelu) * c2 + spc[2] * specf;
}

// ---------------------------------------------------------------------------
// Kernel 3: per-pixel resolve. Only k==0 contributes to the output.
// Stage the strided pixel attributes (pix_to_face stride 32B, bary stride 96B,
// both 16B-aligned per pixel) into LDS via CDNA5 async global->LDS loads,
// drained with s_wait_asynccnt; each lane consumes only its own LDS slot so
// the per-wave ASYNCcnt wait is the only synchronization needed.
// ---------------------------------------------------------------------------
__global__ __launch_bounds__(256) void shade_pixels_kernel(
    const int* __restrict__ pix_to_face, const float* __restrict__ bary,
    const int* __restrict__ faces, const float* __restrict__ vcol,
    const float* __restrict__ bg, float4* __restrict__ out)
{
    __shared__ int    s_face[256];
    __shared__ float4 s_bary[256];

    int t = threadIdx.x;
    unsigned p = blockIdx.x * 256u + t;     // p < 2^20, fits easily

    const int*   gp = pix_to_face + (size_t)p * K_N;        // k = 0 entry
    const float* gb = bary        + (size_t)p * (K_N * 3);  // k = 0, 3 floats (16B load ok: stride 96B)

#if defined(__gfx1250__) && __has_builtin(__builtin_amdgcn_global_load_async_to_lds_b32) && \
    __has_builtin(__builtin_amdgcn_global_load_async_to_lds_b128)
    __builtin_amdgcn_global_load_async_to_lds_b32(
        (as1_i32)gp, (as3_i32)&s_face[t], 0, 0);
    __builtin_amdgcn_global_load_async_to_lds_b128(
        (as1_v4i)gb, (as3_v4i)&s_bary[t], 0, 0);
    asm volatile("s_wait_asynccnt 0" ::: "memory");
#else
    s_face[t] = gp[0];
    s_bary[t] = *(const float4*)gb;
#endif

    int    f = s_face[t];
    float4 w = s_bary[t];

    float r, g, b, a;
    if (f < 0) {
        r = bg[0]; g = bg[1]; b = bg[2]; a = 0.0f;
    } else {
        int i0 = faces[3 * f + 0], i1 = faces[3 * f + 1], i2 = faces[3 * f + 2];
        r = w.x * vcol[3 * i0 + 0] + w.y * vcol[3 * i1 + 0] + w.z * vcol[3 * i2 + 0];
        g = w.x * vcol[3 * i0 + 1] + w.y * vcol[3 * i1 + 1] + w.z * vcol[3 * i2 + 1];
        b = w.x * vcol[3 * i0 + 2] + w.y * vcol[3 * i1 + 2] + w.z * vcol[3 * i2 + 2];
        a = 1.0f;
    }
    out[p] = make_float4(r, g, b, a);
}

// ---------------------------------------------------------------------------
extern "C" void kernel_launch(void* const* d_in, const int* in_sizes, int n_in,
                              void* d_out, int out_size, void* d_ws, size_t ws_size,
                              hipStream_t stream) {
    (void)in_sizes; (void)n_in; (void)out_size; (void)ws_size;

    const float* verts       = (const float*)d_in[0];
    const int*   faces       = (const int*)  d_in[1];
    const float* vert_colors = (const float*)d_in[2];
    const int*   pix_to_face = (const int*)  d_in[3];
    const float* bary        = (const float*)d_in[4];
    const float* light_pos   = (const float*)d_in[5];
    const float* cam_pos     = (const float*)d_in[6];
    const float* ambient     = (const float*)d_in[7];
    const float* diffuse     = (const float*)d_in[8];
    const float* specular    = (const float*)d_in[9];
    const float* background  = (const float*)d_in[10];

    // workspace layout: vn (V*3 floats) | vcol (V*3 floats)  => 2.4 MB
    float* vn   = (float*)d_ws;
    float* vcol = vn + (size_t)V_N * 3;

    zero_kernel<<<(V_N * 3 + 255) / 256, 256, 0, stream>>>(vn, V_N * 3);

    face_normals_kernel<<<(F_N + 255) / 256, 256, 0, stream>>>(verts, faces, vn);

    shade_vertices_kernel<<<(V_N + 255) / 256, 256, 0, stream>>>(
        verts, vert_colors, vn, light_pos, cam_pos, ambient, diffuse, specular, vcol);

    shade_pixels_kernel<<<NPIX / 256, 256, 0, stream>>>(
        pix_to_face, bary, faces, vcol, background, (float4*)d_out);
}